// EDDFNModel_30193620091238
// MI455X (gfx1250) — compile-verified
//
#include <hip/hip_runtime.h>

typedef __attribute__((ext_vector_type(2))) float v2f;
typedef __attribute__((ext_vector_type(8))) float v8f;

#define B_N 2048
#define E_DIM 768
#define H_DIM 384
#define L_SEQ 256
#define N_DOM 9
#define NCHUNK_SPEC 137   // max 16-row chunks after per-category padding: 2048/16 + 9

// ---------------------------------------------------------------------------
// Kernel 1: fused mask-attention pooling. One block (256 thr = 8 waves) per
// sample. Phase 1: each wave computes 32 token scores (lane-strided dot over
// E=768, shfl reduction). Phase 2: block softmax in LDS. Phase 3: weighted sum
// over tokens, thread t owns e = {t, t+256, t+512} (coalesced re-read; row is
// hot in L2).
// ---------------------------------------------------------------------------
__global__ __launch_bounds__(256) void attn_pool_kernel(
    const float* __restrict__ bert, const int* __restrict__ masks,
    const float* __restrict__ att_w, const float* __restrict__ att_b,
    float* __restrict__ pooled)
{
  __shared__ float aw[E_DIM];
  __shared__ float sc[L_SEQ];
  __shared__ float redm[8];
  __shared__ float reds[8];
  __shared__ float s_max, s_sum;

  const int b = blockIdx.x;
  const int t = threadIdx.x;
  const int wave = t >> 5;
  const int lane = t & 31;
  const float* bb = bert + (size_t)b * L_SEQ * E_DIM;

  for (int i = t; i < E_DIM; i += 256) aw[i] = att_w[i];
  __syncthreads();

  for (int j = 0; j < 32; ++j) {
    int l = wave * 32 + j;
    const float* row = bb + (size_t)l * E_DIM;
    float p = 0.f;
    for (int e = lane; e < E_DIM; e += 32) p += row[e] * aw[e];
    for (int o = 16; o > 0; o >>= 1) p += __shfl_xor(p, o, 32);
    if (lane == 0) {
      float s = p + att_b[0];
      if (masks[b * L_SEQ + l] == 0) s = -1e9f;
      sc[l] = s;
    }
  }
  __syncthreads();

  float v = sc[t];
  float m = v;
  for (int o = 16; o > 0; o >>= 1) m = fmaxf(m, __shfl_xor(m, o, 32));
  if (lane == 0) redm[wave] = m;
  __syncthreads();
  if (t == 0) {
    float mm = redm[0];
    for (int i = 1; i < 8; ++i) mm = fmaxf(mm, redm[i]);
    s_max = mm;
  }
  __syncthreads();
  float ex = __expf(v - s_max);
  float s = ex;
  for (int o = 16; o > 0; o >>= 1) s += __shfl_xor(s, o, 32);
  if (lane == 0) reds[wave] = s;
  __syncthreads();
  if (t == 0) {
    float ss = 0.f;
    for (int i = 0; i < 8; ++i) ss += reds[i];
    s_sum = ss;
  }
  __syncthreads();
  sc[t] = ex / s_sum;
  __syncthreads();

  float a0 = 0.f, a1 = 0.f, a2 = 0.f;
  for (int l = 0; l < L_SEQ; ++l) {
    float w = sc[l];
    const float* row = bb + (size_t)l * E_DIM;
    a0 += w * row[t];
    a1 += w * row[t + 256];
    a2 += w * row[t + 512];
  }
  float* po = pooled + (size_t)b * E_DIM;
  po[t] = a0; po[t + 256] = a1; po[t + 512] = a2;
}

// ---------------------------------------------------------------------------
// Kernel 2: route samples by category into 16-row chunks padded per category
// so every WMMA tile uses a single expert matrix. perm[] holds sample indices
// (-1 = pad), chunk_expert[] maps chunk -> expert id. Single block.
// ---------------------------------------------------------------------------
__global__ void route_kernel(const int* __restrict__ cat,
                             int* __restrict__ perm,
                             int* __restrict__ chunk_expert)
{
  __shared__ int cnt[N_DOM];
  __shared__ int base[N_DOM];
  __shared__ int fill[N_DOM];
  int t = threadIdx.x;
  if (t < N_DOM) { cnt[t] = 0; fill[t] = 0; }
  __syncthreads();
  for (int b = t; b < B_N; b += 256) atomicAdd(&cnt[cat[b]], 1);
  __syncthreads();
  if (t == 0) {
    int off = 0;
    for (int d = 0; d < N_DOM; ++d) {
      base[d] = off;
      off += ((cnt[d] + 15) >> 4) << 4;   // pad each segment to 16 rows
    }
  }
  __syncthreads();
  for (int i = t; i < NCHUNK_SPEC * 16; i += 256) perm[i] = -1;
  for (int c = t; c < NCHUNK_SPEC; c += 256) {
    int e = 0;
    for (int d = 0; d < N_DOM; ++d) if (c * 16 >= base[d]) e = d;
    chunk_expert[c] = e;
  }
  __syncthreads();
  for (int b = t; b < B_N; b += 256) {
    int d = cat[b];
    int p = atomicAdd(&fill[d], 1);
    perm[base[d] + p] = b;
  }
}

// ---------------------------------------------------------------------------
// Kernel 3: generic FP32 WMMA GEMM  C = act(A[M,K] @ B[K,N] + bias).
// Block = 128 thr (4 waves); block tile = 16 (M) x 64*NT (N); wave w owns the
// 16 x 16*NT strip (NT accumulators share one A fragment -> half the LDS
// A-traffic at NT=2). K consumed in LDS-staged chunks of 64 -> 16*NT
// V_WMMA_F32_16X16X4_F32 per chunk.
// B staging uses CDNA5 async global->LDS DMA (GLOBAL_LOAD_ASYNC_TO_LDS_B128,
// ASYNCcnt); row stride padded to a 16B multiple so B128 writes stay aligned.
// A staging stays on the sync path because gather pads (-1) must write zeros.
// A-frag layout (16x4 f32): lanes 0-15 K={0,1}, lanes 16-31 K={2,3}.
// D layout: VGPR j = row (j + 8*(lane>=16)), col = lane&15.
// ---------------------------------------------------------------------------
template<int NT>
__global__ __launch_bounds__(128) void gemm16_wmma(
    const float* __restrict__ A, int lda,
    const int* __restrict__ rowidx,
    const int* __restrict__ chunk_expert,
    const float* __restrict__ Bm, long b_stride, int ldb,
    const float* __restrict__ bias, long bias_stride,
    float* __restrict__ C, int ldc, int ncol_off,
    int K, int relu)
{
  constexpr int NB = 64 * NT;           // block N span
  constexpr int BSTR = NB + 4;          // padded row stride (16B multiple)
  __shared__ float As[16][68];          // +4 pad breaks 16-way bank conflict
  __shared__ float Bs[64][BSTR];

  const int t = threadIdx.x;
  const int chunk = blockIdx.x;
  const int n0 = blockIdx.y * NB;
  const int wave = t >> 5;
  const int lane = t & 31;
  const int half = lane >> 4;
  const int lr = lane & 15;

  const int expert = chunk_expert ? chunk_expert[chunk] : 0;
  const float* Bp = Bm + (long)expert * b_stride;

  v8f acc[NT];
#pragma unroll
  for (int j = 0; j < NT; ++j) acc[j] = (v8f){};

  const int kchunks = K >> 6;
  for (int kc = 0; kc < kchunks; ++kc) {
    const int kbase = kc << 6;
    // ---- B: 64 x NB chunk via async global->LDS DMA (b128 = 4 floats/lane)
#pragma unroll
    for (int i = 0; i < NB / 8; ++i) {          // (64*NB/4)/128 iters
      int idx4 = t + i * 128;
      int kk = idx4 / (NB / 4);
      int nn = (idx4 % (NB / 4)) * 4;
      unsigned ldsa = (unsigned)(size_t)&Bs[kk][nn];
      const float* g = Bp + (size_t)(kbase + kk) * ldb + n0 + nn;
      asm volatile("global_load_async_to_lds_b128 %0, %1, off"
                   :: "v"(ldsa), "v"(g) : "memory");
    }
    // ---- A: 16x64 chunk (sync path; gather pads write zeros)
#pragma unroll
    for (int i = 0; i < 8; ++i) {
      int idx = t + i * 128;
      int r = idx >> 6, kk = idx & 63;
      int rg = rowidx ? rowidx[chunk * 16 + r] : chunk * 16 + r;
      As[r][kk] = (rg < 0) ? 0.f : A[(long)rg * lda + kbase + kk];
    }
    asm volatile("s_wait_asynccnt 0x0" ::: "memory");
    __syncthreads();

#pragma unroll
    for (int kk = 0; kk < 64; kk += 4) {
      int ka = kk + 2 * half;
      v2f a;
      a.x = As[lr][ka];
      a.y = As[lr][ka + 1];
#pragma unroll
      for (int j = 0; j < NT; ++j) {
        int col = wave * 16 * NT + j * 16 + lr;
        v2f b;
        b.x = Bs[ka][col];
        b.y = Bs[ka + 1][col];
        acc[j] = __builtin_amdgcn_wmma_f32_16x16x4_f32(
            /*neg_a=*/false, a, /*neg_b=*/false, b,
            /*c_mod=*/(short)0, acc[j], /*reuse_a=*/false, /*reuse_b=*/false);
      }
    }
    __syncthreads();
  }

#pragma unroll
  for (int j = 0; j < NT; ++j) {
    const int ncol = n0 + wave * 16 * NT + j * 16 + lr;
    const float bv = bias ? bias[(long)expert * bias_stride + ncol] : 0.f;
#pragma unroll
    for (int q = 0; q < 8; ++q) {
      int mrow = q + 8 * half;
      int rg = rowidx ? rowidx[chunk * 16 + mrow] : chunk * 16 + mrow;
      if (rg >= 0) {
        float vv = acc[j][q] + bv;
        if (relu) vv = fmaxf(vv, 0.f);
        C[(long)rg * ldc + ncol_off + ncol] = vv;
      }
    }
  }
}

// ---------------------------------------------------------------------------
// Kernel 4: classifier  out[b] = sigmoid(feature[b,:768] . Wc + bc). Wave/row.
// ---------------------------------------------------------------------------
__global__ __launch_bounds__(256) void cls_kernel(
    const float* __restrict__ feat, const float* __restrict__ Wc,
    const float* __restrict__ bc, float* __restrict__ out)
{
  int gw = (blockIdx.x * blockDim.x + threadIdx.x) >> 5;
  int lane = threadIdx.x & 31;
  if (gw >= B_N) return;
  const float* f = feat + (size_t)gw * (2 * H_DIM);
  float p = 0.f;
  for (int i = lane; i < 2 * H_DIM; i += 32) p += f[i] * Wc[i];
  for (int o = 16; o > 0; o >>= 1) p += __shfl_xor(p, o, 32);
  if (lane == 0) {
    float x = p + bc[0];
    out[gw] = 1.f / (1.f + __expf(-x));
  }
}

// ---------------------------------------------------------------------------
// Kernel 5: domain head  dp[b,d] = hdo[b,:] . Wdo2[:,d] + bdo2[d].  N=9, tiny.
// ---------------------------------------------------------------------------
__global__ __launch_bounds__(256) void dom_kernel(
    const float* __restrict__ hdo, const float* __restrict__ W2,
    const float* __restrict__ b2, float* __restrict__ dp)
{
  int i = blockIdx.x * blockDim.x + threadIdx.x;
  if (i >= B_N * N_DOM) return;
  int b = i / N_DOM, d = i % N_DOM;
  const float* h = hdo + (size_t)b * H_DIM;
  float s = b2[d];
  for (int k = 0; k < H_DIM; ++k) s += h[k] * W2[k * N_DOM + d];
  dp[i] = s;
}

// ---------------------------------------------------------------------------
extern "C" void kernel_launch(void* const* d_in, const int* in_sizes, int n_in,
                              void* d_out, int out_size, void* d_ws, size_t ws_size,
                              hipStream_t stream)
{
  (void)in_sizes; (void)n_in; (void)out_size; (void)ws_size;

  const float* bert  = (const float*)d_in[0];
  const int*   masks = (const int*)d_in[1];
  const int*   cat   = (const int*)d_in[2];
  /* d_in[3] = alpha: identity in forward */
  const float* att_w = (const float*)d_in[4];
  const float* att_b = (const float*)d_in[5];
  const float* Wsh   = (const float*)d_in[6];
  const float* bsh   = (const float*)d_in[7];
  const float* Wsp   = (const float*)d_in[8];
  const float* bsp   = (const float*)d_in[9];
  const float* Wd1   = (const float*)d_in[10];
  const float* bd1   = (const float*)d_in[11];
  const float* Wd2   = (const float*)d_in[12];
  const float* bd2   = (const float*)d_in[13];
  const float* Wc    = (const float*)d_in[14];
  const float* bc    = (const float*)d_in[15];
  const float* Wdo1  = (const float*)d_in[16];
  const float* bdo1  = (const float*)d_in[17];
  const float* Wdo2  = (const float*)d_in[18];
  const float* bdo2  = (const float*)d_in[19];

  // outputs concatenated flat: out[B], rec[B,E], pooled[B,E], domain_pred[B,9]
  float* out_cls = (float*)d_out;
  float* rec     = out_cls + B_N;
  float* pooled  = rec + (size_t)B_N * E_DIM;
  float* dompred = pooled + (size_t)B_N * E_DIM;

  // workspace carve-up (~10 MB)
  char* ws = (char*)d_ws;
  float* feature = (float*)ws;  ws += (size_t)B_N * 2 * H_DIM * sizeof(float);
  float* t1      = (float*)ws;  ws += (size_t)B_N * 64 * sizeof(float);
  float* hdo     = (float*)ws;  ws += (size_t)B_N * H_DIM * sizeof(float);
  int*   perm    = (int*)ws;    ws += (size_t)NCHUNK_SPEC * 16 * sizeof(int);
  int*   cexp    = (int*)ws;    ws += (size_t)(NCHUNK_SPEC + 1) * sizeof(int);

  attn_pool_kernel<<<B_N, 256, 0, stream>>>(bert, masks, att_w, att_b, pooled);
  route_kernel<<<1, 256, 0, stream>>>(cat, perm, cexp);

  // shared = relu(pooled @ Wsh + bsh)            -> feature[:, 0:384]
  gemm16_wmma<2><<<dim3(B_N / 16, H_DIM / 128), 128, 0, stream>>>(
      pooled, E_DIM, nullptr, nullptr, Wsh, 0, H_DIM, bsh, 0,
      feature, 2 * H_DIM, 0, E_DIM, 1);
  // specific = relu(pooled @ Wsp[cat] + bsp[cat]) -> feature[:, 384:768]
  gemm16_wmma<2><<<dim3(NCHUNK_SPEC, H_DIM / 128), 128, 0, stream>>>(
      pooled, E_DIM, perm, cexp, Wsp, (long)E_DIM * H_DIM, H_DIM, bsp, H_DIM,
      feature, 2 * H_DIM, H_DIM, E_DIM, 1);
  // t1 = relu(feature @ Wd1 + bd1)   (N=64 -> NT=1)
  gemm16_wmma<1><<<dim3(B_N / 16, 1), 128, 0, stream>>>(
      feature, 2 * H_DIM, nullptr, nullptr, Wd1, 0, 64, bd1, 0,
      t1, 64, 0, 2 * H_DIM, 1);
  // rec = relu(t1 @ Wd2 + bd2)
  gemm16_wmma<2><<<dim3(B_N / 16, E_DIM / 128), 128, 0, stream>>>(
      t1, 64, nullptr, nullptr, Wd2, 0, E_DIM, bd2, 0,
      rec, E_DIM, 0, 64, 1);
  // hdo = relu(shared @ Wdo1 + bdo1)   (shared = feature[:, :384], lda=768)
  gemm16_wmma<2><<<dim3(B_N / 16, H_DIM / 128), 128, 0, stream>>>(
      feature, 2 * H_DIM, nullptr, nullptr, Wdo1, 0, H_DIM, bdo1, 0,
      hdo, H_DIM, 0, H_DIM, 1);

  cls_kernel<<<B_N / 8, 256, 0, stream>>>(feature, Wc, bc, out_cls);
  dom_kernel<<<(B_N * N_DOM + 255) / 256, 256, 0, stream>>>(hdo, Wdo2, bdo2, dompred);
}